// SpatialGCNLayer_70626442215849
// MI455X (gfx1250) — compile-verified
//
#include <hip/hip_runtime.h>
#include <hip/hip_bf16.h>

// Shapes fixed by the reference setup.
#define BB 2
#define TT 12
#define NN 10000
#define CIN 64
#define COUT 32
#define KK1 17
#define HH 4

typedef float v2f __attribute__((ext_vector_type(2)));
typedef float v8f __attribute__((ext_vector_type(8)));

// ---------------------------------------------------------------------------
// Kernel 1: projection GEMM  proj[m,o] = sum_i x[m,i]*W[o,i] + b[o]
// via V_WMMA_F32_16X16X4_F32. One wave -> one 16(M) x 32(N) tile
// (two 16x16 accumulators). K=64 -> 16 steps of K=4.
//
// A (16x4 f32) layout per ISA: VGPR0 = K0 (lanes 0-15) / K2 (lanes 16-31),
// VGPR1 = K1 / K3  =>  lane loads float2 at column (4*ks + 2*half).
// B (4x16 f32) mirrors that with N striped across lanes within a VGPR,
// which for row-major W[o,i] is again a float2 load at the same column.
// ---------------------------------------------------------------------------
__global__ __launch_bounds__(256) void proj_wmma_kernel(
    const float* __restrict__ x, const float* __restrict__ W,
    const float* __restrict__ bias, float* __restrict__ proj, int M) {
  const int lane = threadIdx.x & 31;
  const int half = lane >> 4;   // 0: lanes 0-15, 1: lanes 16-31
  const int l15  = lane & 15;
  const long wave = (long)blockIdx.x * (blockDim.x >> 5) + (threadIdx.x >> 5);
  const long row0 = wave * 16;            // M-tile base (wave uniform)
  if (row0 >= M) return;                  // wave-uniform: EXEC stays all-ones

  const float* ap  = x + (row0 + l15) * CIN + half * 2;       // A row
  const float* bp0 = W + (long)l15        * CIN + half * 2;   // N-tile 0 (o=l15)
  const float* bp1 = W + (long)(l15 + 16) * CIN + half * 2;   // N-tile 1 (o=16+l15)

  v8f acc0 = {};
  v8f acc1 = {};
#pragma unroll
  for (int ks = 0; ks < 16; ++ks) {
    v2f a  = *(const v2f*)(ap  + ks * 4);
    v2f b0 = *(const v2f*)(bp0 + ks * 4);
    v2f b1 = *(const v2f*)(bp1 + ks * 4);
    // 8 args: (neg_a, A, neg_b, B, c_mod, C, reuse_a, reuse_b)
    acc0 = __builtin_amdgcn_wmma_f32_16x16x4_f32(false, a, false, b0,
                                                 (short)0, acc0, false, false);
    acc1 = __builtin_amdgcn_wmma_f32_16x16x4_f32(false, a, false, b1,
                                                 (short)0, acc1, false, false);
  }

  const float bv0 = bias[l15];
  const float bv1 = bias[l15 + 16];
  // C/D layout: VGPR r -> M=r (lanes 0-15) / M=r+8 (lanes 16-31), N = lane&15.
#pragma unroll
  for (int r = 0; r < 8; ++r) {
    const long m = row0 + r + half * 8;
    proj[m * COUT + l15]      = acc0[r] + bv0;   // lanes 0-15: 64B contiguous
    proj[m * COUT + 16 + l15] = acc1[r] + bv1;
  }
}

// ---------------------------------------------------------------------------
// Kernel 2: per-node Gaussian weights + distance aggregate.
// One thread per node n; loops K1 neighbors x H heads.
// ---------------------------------------------------------------------------
__global__ __launch_bounds__(256) void weights_kernel(
    const float* __restrict__ nd, const int* __restrict__ nn,
    float* __restrict__ wbuf, float* __restrict__ dagg) {
  const int n = blockIdx.x * blockDim.x + threadIdx.x;
  if (n >= NN) return;
  float da[HH] = {0.f, 0.f, 0.f, 0.f};
  for (int k = 0; k < KK1; ++k) {
    const float d   = nd[n * KK1 + k];
    const int node  = nn[n * KK1 + k];
    const float d2  = d * d;
    const float dc  = isinf(d) ? 0.f : d;
#pragma unroll
    for (int h = 0; h < HH; ++h) {
      const float lam = (float)(h + 1) * (1.0f / HH);   // SIGMA == 1
      float wv = __expf(-d2 * lam);
      if (node == -1)   wv = 0.f;
      if (wv < 1e-5f)   wv = 0.f;
      wbuf[(n * KK1 + k) * HH + h] = wv;
      da[h] += wv * dc;
    }
  }
#pragma unroll
  for (int h = 0; h < HH; ++h) dagg[n * HH + h] = da[h];
}

// ---------------------------------------------------------------------------
// Kernel 3: gather + head-weighted aggregate (bandwidth-critical).
// One wave per (b,t,n); lane = output channel c. 17 coalesced 128B gathers
// from the L2-resident proj slice, 4 coalesced 128B stores.
// ---------------------------------------------------------------------------
__global__ __launch_bounds__(256) void agg_kernel(
    const float* __restrict__ proj, const float* __restrict__ wbuf,
    const float* __restrict__ dagg, const int* __restrict__ nn,
    float* __restrict__ out) {
  const int lane = threadIdx.x & 31;  // channel c (COUT == warpSize == 32)
  const long wave = (long)blockIdx.x * (blockDim.x >> 5) + (threadIdx.x >> 5);
  const long total = (long)BB * TT * NN;
  if (wave >= total) return;
  const int n   = (int)(wave % NN);
  const long bt = wave / NN;
  const float* pslice = proj + bt * (long)NN * COUT;

  float acc0 = 0.f, acc1 = 0.f, acc2 = 0.f, acc3 = 0.f;
  for (int k = 0; k < KK1; ++k) {
    const int node = nn[n * KK1 + k];      // wave-uniform
    float val = 0.f;
    if (node >= 0 && node < NN)            // -1 / pad row -> zeros
      val = pslice[(long)node * COUT + lane];
    const float4 w4 = *(const float4*)(wbuf + (n * KK1 + k) * HH);
    acc0 += w4.x * val;
    acc1 += w4.y * val;
    acc2 += w4.z * val;
    acc3 += w4.w * val;
  }

  float* o = out + wave * (long)HH * COUT;
  o[0 * COUT + lane] = acc0 + dagg[n * HH + 0];
  o[1 * COUT + lane] = acc1 + dagg[n * HH + 1];
  o[2 * COUT + lane] = acc2 + dagg[n * HH + 2];
  o[3 * COUT + lane] = acc3 + dagg[n * HH + 3];
}

// ---------------------------------------------------------------------------
extern "C" void kernel_launch(void* const* d_in, const int* in_sizes, int n_in,
                              void* d_out, int out_size, void* d_ws, size_t ws_size,
                              hipStream_t stream) {
  const float* x  = (const float*)d_in[0];  // (B,T,N,CIN)
  const float* W  = (const float*)d_in[1];  // (COUT,CIN)
  const float* b  = (const float*)d_in[2];  // (COUT,)
  const float* nd = (const float*)d_in[3];  // (N,K1)
  const int*   nn = (const int*)d_in[4];    // (N,K1)
  float* out = (float*)d_out;               // (B,T,N,H,COUT)

  const long M = (long)BB * TT * NN;        // 240000 rows (divisible by 16*8)

  // Workspace layout (floats): proj[M*COUT] | w[N*K1*H] | dagg[N*H]  (~33.6 MB)
  float* proj = (float*)d_ws;
  float* wbuf = proj + M * COUT;
  float* dagg = wbuf + (long)NN * KK1 * HH;

  // 1) Projection GEMM via WMMA: 15000 M-tiles, 8 waves/block.
  const int tiles = (int)(M / 16);
  proj_wmma_kernel<<<(tiles + 7) / 8, 256, 0, stream>>>(x, W, b, proj, (int)M);

  // 2) Weights + distance aggregate.
  weights_kernel<<<(NN + 255) / 256, 256, 0, stream>>>(nd, nn, wbuf, dagg);

  // 3) Gather + aggregate: one wave per (b,t,n).
  agg_kernel<<<(int)((M + 7) / 8), 256, 0, stream>>>(proj, wbuf, dagg, nn, out);
}